// CrowdHumanPostProcess_77249281786084
// MI455X (gfx1250) — compile-verified
//
#include <hip/hip_runtime.h>
#include <cstdint>
#include <cstddef>

typedef float v2f __attribute__((ext_vector_type(2)));
typedef float v8f __attribute__((ext_vector_type(8)));
typedef unsigned long long u64;

#define THREADS_SORT 1024
#define RADIX 256

// ---- order-preserving key transform (descending float order) ----
__device__ __forceinline__ unsigned fwd_key(float f) {
  unsigned b = __float_as_uint(f);
  unsigned asc = (b & 0x80000000u) ? ~b : (b | 0x80000000u); // ascending sortable
  return ~asc;                                               // descending
}
__device__ __forceinline__ float inv_key(unsigned K) {
  unsigned asc = ~K;
  unsigned b = (asc & 0x80000000u) ? (asc ^ 0x80000000u) : ~asc;
  return __uint_as_float(b);
}

__device__ __forceinline__ unsigned ballot32(bool p) {
  return __builtin_amdgcn_ballot_w32(p);  // wave32 on gfx1250
}

// ---- K0: build (key<<32)|index pairs, fused with pass-0 tile histogram ----
__global__ void k_init_hist(const float* __restrict__ logits,
                            u64* __restrict__ dst,
                            unsigned* __restrict__ tileHist, int N, int NT) {
  __shared__ unsigned h[RADIX];
  int b = blockIdx.y, tile = blockIdx.x, tid = threadIdx.x;
  if (tid < RADIX) h[tid] = 0;
  int e = tile * THREADS_SORT + tid;
  if (e < N) __builtin_prefetch(&logits[(size_t)b * N + e], 0, 0);
  __syncthreads();
  if (e < N) {
    size_t off = (size_t)b * N + e;
    unsigned K = fwd_key(logits[off]);
    dst[off] = ((u64)K << 32) | (unsigned)e;
    atomicAdd(&h[K & 0xFFu], 1u); // pass-0 digit
  }
  __syncthreads();
  if (tid < RADIX) tileHist[(size_t)(b * NT + tile) * RADIX + tid] = h[tid];
}

// ---- K1: per-tile 256-bin histogram (passes 1..3) ----
__global__ void k_hist(const u64* __restrict__ src,
                       unsigned* __restrict__ tileHist, int N, int NT, int shift) {
  __shared__ unsigned h[RADIX];
  int b = blockIdx.y, tile = blockIdx.x, tid = threadIdx.x;
  int e = tile * THREADS_SORT + tid;
  if (e < N) __builtin_prefetch(&src[(size_t)b * N + e], 0, 0); // global_prefetch_b8
  if (tid < RADIX) h[tid] = 0;
  __syncthreads();
  if (e < N) {
    unsigned d = (unsigned)(src[(size_t)b * N + e] >> (32 + shift)) & 0xFFu;
    atomicAdd(&h[d], 1u);
  }
  __syncthreads();
  if (tid < RADIX) tileHist[(size_t)(b * NT + tile) * RADIX + tid] = h[tid];
}

// ---- K2: per-segment digit scan + per-tile exclusive offsets (in place) ----
__global__ void k_scan(unsigned* __restrict__ tileHist, int NT) {
  __shared__ unsigned tot[RADIX];
  int b = blockIdx.x, d = threadIdx.x; // 256 threads
  size_t base = (size_t)b * NT * RADIX + d;
  unsigned t = 0;
  for (int tl = 0; tl < NT; ++tl) t += tileHist[base + (size_t)tl * RADIX];
  tot[d] = t;
  __syncthreads();
  unsigned val = t; // inclusive scan over digits (Hillis-Steele)
  for (int off = 1; off < RADIX; off <<= 1) {
    unsigned nb = (d >= off) ? tot[d - off] : 0u;
    __syncthreads();
    val += nb;
    tot[d] = val;
    __syncthreads();
  }
  unsigned running = val - t; // exclusive digit base within segment
  for (int tl = 0; tl < NT; ++tl) {
    size_t i = base + (size_t)tl * RADIX;
    unsigned v = tileHist[i];
    tileHist[i] = running;
    running += v;
  }
}

// ---- K3: stable scatter (wave32 digit match via ballot + cross-wave prefix) ----
__global__ void k_scatter(const u64* __restrict__ src,
                          u64* __restrict__ dst,
                          const unsigned* __restrict__ tileOff,
                          int N, int NT, int shift) {
  __shared__ unsigned wdc[32 * RADIX]; // [wave][digit]
  int b = blockIdx.y, tile = blockIdx.x, tid = threadIdx.x;
  unsigned lane = (unsigned)tid & 31u, wave = (unsigned)tid >> 5;
  int e = tile * THREADS_SORT + tid;
  bool valid = (e < N);
  u64 kv = 0;
  if (valid) kv = src[(size_t)b * N + e];
  for (int i = tid; i < 32 * RADIX; i += THREADS_SORT) wdc[i] = 0;
  __syncthreads();
  unsigned digit = (unsigned)(kv >> (32 + shift)) & 0xFFu;
  unsigned peers = ballot32(valid); // match only among valid lanes
#pragma unroll
  for (int bit = 0; bit < 8; ++bit) {
    unsigned bb = ballot32(((digit >> bit) & 1u) != 0u);
    peers &= ((digit >> bit) & 1u) ? bb : ~bb;
  }
  unsigned ltmask = (1u << lane) - 1u;
  unsigned rank = __popc(peers & ltmask);
  unsigned cnt = __popc(peers);
  if (valid && rank == 0u) wdc[wave * RADIX + digit] = cnt; // one leader per group
  __syncthreads();
  if (tid < RADIX) { // exclusive prefix across 32 waves per digit
    unsigned running = 0;
#pragma unroll 4
    for (int w = 0; w < 32; ++w) {
      unsigned v = wdc[w * RADIX + tid];
      wdc[w * RADIX + tid] = running;
      running += v;
    }
  }
  __syncthreads();
  if (valid) {
    unsigned pos = tileOff[(size_t)(b * NT + tile) * RADIX + digit] +
                   wdc[wave * RADIX + digit] + rank;
    dst[(size_t)b * N + pos] = kv;
  }
}

// ---- fused epilogue: scores + labels + WMMA box transform ----
// D(16x16) = A(16 boxes x 4 coords) x Bmat(4 x 16), [w,h,w,h] folded into Bmat.
// Per wave: 16 sorted entries -> scores/labels (lanes 0-15) + boxes via
// v_wmma_f32_16x16x4_f32, then intra-wave LDS transpose -> coalesced b64 stores.
__global__ void k_epilogue(const u64* __restrict__ sorted,
                           const float* __restrict__ boxes,
                           const int* __restrict__ tsizes,
                           float* __restrict__ out,       // scores | labels | boxes
                           int N, long long BN, int tiles16, long long totalTasks) {
  __shared__ float stage[8][64]; // per-wave [coord*16 + box], 8 waves/block
  int wlocal = threadIdx.x >> 5;
  long long wtask = (long long)blockIdx.x * 8 + wlocal;
  if (wtask >= totalTasks) return; // wave-uniform: EXEC all-ones at the WMMA
  int b = (int)(wtask / tiles16);
  int base = (int)(wtask - (long long)b * tiles16) * 16;
  unsigned lane = (unsigned)threadIdx.x & 31u;
  unsigned row = lane & 15u;   // A: M row (box in tile) / B,D: N column
  unsigned half = lane >> 4;   // K split: lanes 0-15 -> K{0,1}, 16-31 -> K{2,3}
  bool fullTile = (base + 16 <= N);
  float W = (float)tsizes[b * 2 + 1];
  float H = (float)tsizes[b * 2 + 0];

  int r0 = base + (int)row;
  u64 kv = 0;
  if (fullTile) kv = sorted[(size_t)b * N + r0];
  else if (r0 < N) kv = sorted[(size_t)b * N + r0];
  unsigned sidx = (unsigned)(kv & 0xFFFFFFFFull);

  // scores + labels from lanes 0-15 (sorted keys invert back to logits)
  if (half == 0u && r0 < N) {
    size_t soff = (size_t)b * N + r0;
    float f = inv_key((unsigned)(kv >> 32));
    out[soff] = 1.0f / (1.0f + __expf(-f)); // sigmoid of sorted logit
    out[(size_t)BN + soff] = 1.0f;          // labels: transform[idx % 1] == 1
  }

  const float* bp = boxes + ((size_t)b * N + sidx) * 4 + half * 2;
  v2f A; A.x = bp[0]; A.y = bp[1]; // A[row][k0], A[row][k0+1]

  unsigned col = row, k0 = half * 2;
  auto coef = [&](unsigned k, unsigned j) -> float { // Bmat[k][j], scale folded in
    if (j >= 4) return 0.0f;
    float s = (j & 1u) ? H : W;
    if ((j & 1u) == 0u) { // x outputs: cx(k=0), w(k=2)
      if (k == 0) return s;
      if (k == 2) return (j == 0) ? -0.5f * s : 0.5f * s;
    } else {              // y outputs: cy(k=1), h(k=3)
      if (k == 1) return s;
      if (k == 3) return (j == 1) ? -0.5f * s : 0.5f * s;
    }
    return 0.0f;
  };
  v2f Bv; Bv.x = coef(k0, col); Bv.y = coef(k0 + 1, col);
  v8f C = {};
  v8f D = __builtin_amdgcn_wmma_f32_16x16x4_f32(
      false, A, false, Bv, (short)0, C, false, false);

  // stage D into LDS col-major: lane (col<4) holds D[r] = box (r + half*8), coord col
  if (col < 4u) {
    float* sp = &stage[wlocal][col * 16u + half * 8u];
    *(float4*)(sp)     = make_float4(D[0], D[1], D[2], D[3]);
    *(float4*)(sp + 4) = make_float4(D[4], D[5], D[6], D[7]);
  }
  __builtin_amdgcn_wave_barrier(); // DS ops are in-order within a wave
  // read back box-major: lane L -> output floats 2L, 2L+1 of this 64-float tile
  unsigned m = lane >> 1;          // box within tile
  unsigned j = (lane & 1u) * 2u;   // coordinate pair start
  float o0 = stage[wlocal][j * 16u + m];
  float o1 = stage[wlocal][(j + 1u) * 16u + m];
  float* outBoxes = out + 2 * BN;
  if (fullTile) {
    *(float2*)(outBoxes + ((size_t)b * N + base) * 4 + 2u * lane) =
        make_float2(o0, o1);
  } else {
    int orow = base + (int)m;
    if (orow < N)
      *(float2*)(outBoxes + ((size_t)b * N + orow) * 4 + j) = make_float2(o0, o1);
  }
}

extern "C" void kernel_launch(void* const* d_in, const int* in_sizes, int n_in,
                              void* d_out, int out_size, void* d_ws, size_t ws_size,
                              hipStream_t stream) {
  const float* logits = (const float*)d_in[0];
  const float* boxes  = (const float*)d_in[1];
  const int*   tsizes = (const int*)d_in[2];
  float* out = (float*)d_out;

  const int B = in_sizes[2] / 2;           // 64
  const int N = in_sizes[0] / B;           // 100000 (C == 1)
  const long long BN = (long long)B * N;
  const int NT = (N + THREADS_SORT - 1) / THREADS_SORT; // 98

  // workspace: 2x u64 ping-pong (102.4 MB) + tile histograms (6.4 MB)
  u64* buf0 = (u64*)d_ws;
  u64* buf1 = buf0 + BN;
  unsigned* tileHist = (unsigned*)(buf1 + BN);

  dim3 gridSeg(NT, B);
  // init fused with pass-0 histogram (keys are written in tile order)
  k_init_hist<<<gridSeg, THREADS_SORT, 0, stream>>>(logits, buf0, tileHist, N, NT);

  u64* s = buf0;
  u64* d = buf1;
  for (int p = 0; p < 4; ++p) { // LSD radix, 8-bit digits of the hi word
    int shift = p * 8;
    if (p > 0)
      k_hist<<<gridSeg, THREADS_SORT, 0, stream>>>(s, tileHist, N, NT, shift);
    k_scan   <<<dim3(B), RADIX,        0, stream>>>(tileHist, NT);
    k_scatter<<<gridSeg, THREADS_SORT, 0, stream>>>(s, d, tileHist, N, NT, shift);
    u64* t = s; s = d; d = t;
  }
  // after 4 passes sorted data is back in buf0 (== s)

  int tiles16 = (N + 15) / 16;                    // 6250
  long long tasks = (long long)B * tiles16;       // 400000 wave-tasks
  int blocks = (int)((tasks + 7) / 8);            // 8 waves per 256-thread block
  k_epilogue<<<blocks, 256, 0, stream>>>(s, boxes, tsizes, out, N, BN, tiles16, tasks);
}